// DotPredictor_24352464569641
// MI455X (gfx1250) — compile-verified
//
#include <hip/hip_runtime.h>

typedef __attribute__((ext_vector_type(2))) float v2f;
typedef __attribute__((ext_vector_type(8))) float v8f;

#define D_FEAT 64
#define EDGES_PER_WAVE 16
#define WAVES_PER_BLOCK 8

// Per-edge dot product via WMMA: each wave32 computes 16 edge scores as the
// diagonal of a 16x16 fp32 outer-product tile, accumulated over K=64 with
// sixteen V_WMMA_F32_16X16X4_F32 ops. Memory-bound (h table is L2-resident);
// the 16x compute overshoot is free.
__global__ __launch_bounds__(WAVES_PER_BLOCK * 32)
void edge_dot_wmma_kernel(const float* __restrict__ h,
                          const int* __restrict__ src,
                          const int* __restrict__ dst,
                          float* __restrict__ out,
                          int n_edges)
{
    const int lane     = threadIdx.x & 31;
    const int wave     = threadIdx.x >> 5;
    const int tile     = blockIdx.x * WAVES_PER_BLOCK + wave;
    const int edgeBase = tile * EDGES_PER_WAVE;
    if (edgeBase >= n_edges) return;            // wave-uniform: EXEC stays full

    // Edge owned by this lane's row (A) / column (B) of the WMMA tile.
    int e = edgeBase + (lane & 15);
    if (e >= n_edges) e = n_edges - 1;          // clamp (keeps EXEC all-ones)

    // ISA fragment layout for 32-bit A (16x4) and B (4x16) is identical per
    // lane: lanes 0-15 hold features {4k+0,4k+1}, lanes 16-31 hold {4k+2,4k+3}
    // of edge (lane&15), in VGPRs {0,1} of each K-chunk -> one float2 load.
    const int fOff = (lane >> 4) << 1;

    const int si = src[e];
    const int di = dst[e];
    const float* aPtr = h + si * D_FEAT + fOff;
    const float* bPtr = h + di * D_FEAT + fOff;

    // Issue all 32 fragment loads (b64, immediate offsets) before any wait.
    v2f afrag[16];
    v2f bfrag[16];
#pragma unroll
    for (int kc = 0; kc < 16; ++kc) {
        afrag[kc] = *(const v2f*)(aPtr + kc * 4);
        bfrag[kc] = *(const v2f*)(bPtr + kc * 4);
    }

    // D[i][j] = sum_k h[src_i][k] * h[dst_j][k]; diagonal = edge scores.
    v8f acc = {};
#pragma unroll
    for (int kc = 0; kc < 16; ++kc) {
        acc = __builtin_amdgcn_wmma_f32_16x16x4_f32(
            /*neg_a=*/false, afrag[kc],
            /*neg_b=*/false, bfrag[kc],
            /*c_mod=*/(short)0, acc,
            /*reuse_a=*/false, /*reuse_b=*/false);
    }

    // Diagonal extraction from the C/D layout (VGPR r: lanes 0-15 -> M=r,
    // lanes 16-31 -> M=r+8):
    //   edge i in [0,8):  lane i,    acc[i]
    //   edge i in [8,16): lane i+16, acc[i-8]
    if (lane < 8) {
        int i = lane;
        if (edgeBase + i < n_edges) out[edgeBase + i] = acc[i];
    } else if (lane >= 24) {
        int i = lane - 16;
        if (edgeBase + i < n_edges) out[edgeBase + i] = acc[lane - 24];
    }
}

extern "C" void kernel_launch(void* const* d_in, const int* in_sizes, int n_in,
                              void* d_out, int out_size, void* d_ws, size_t ws_size,
                              hipStream_t stream) {
    const float* h   = (const float*)d_in[0];
    const int*   src = (const int*)d_in[1];
    const int*   dst = (const int*)d_in[2];
    float*       out = (float*)d_out;

    const int n_edges = in_sizes[1];
    const int n_tiles = (n_edges + EDGES_PER_WAVE - 1) / EDGES_PER_WAVE;
    const int blocks  = (n_tiles + WAVES_PER_BLOCK - 1) / WAVES_PER_BLOCK;

    edge_dot_wmma_kernel<<<blocks, WAVES_PER_BLOCK * 32, 0, stream>>>(
        h, src, dst, out, n_edges);
}